// Model_39676907884806
// MI455X (gfx1250) — compile-verified
//
#include <hip/hip_runtime.h>
#include <math.h>

typedef float v2f __attribute__((ext_vector_type(2)));
typedef float v8f __attribute__((ext_vector_type(8)));
typedef int   v4i __attribute__((ext_vector_type(4)));

#define AS1 __attribute__((address_space(1)))
#define AS3 __attribute__((address_space(3)))

typedef AS1 v4i* gv4i_p;   // global 16B vector pointer (builtin param 1)
typedef AS3 v4i* lv4i_p;   // LDS 16B vector pointer   (builtin param 2)

#define Bn 64
#define Hn 48
#define Sn 64
#define Dn 64
#define NTOT (Bn * Hn * Sn * Dn)   // 12,582,912 elements per tensor chunk

#if defined(__gfx1250__) && __has_builtin(__builtin_amdgcn_global_load_async_to_lds_b128)
#define USE_ASYNC_LDS 1
#else
#define USE_ASYNC_LDS 0
#endif

// ---------------------------------------------------------------------------
// Kernel 1: pass-through copies, float4-vectorized.
// out[0..N) = q, out[N..2N) = k, out[2N..3N) = v, out[3N..4N) = mask (->float)
// ---------------------------------------------------------------------------
__global__ __launch_bounds__(256) void copy_passthru_kernel(
    const float4* __restrict__ q4, const float4* __restrict__ k4,
    const float4* __restrict__ v4, const uchar4* __restrict__ m4,
    float4* __restrict__ out4) {
  size_t i = (size_t)blockIdx.x * blockDim.x + threadIdx.x;
  const size_t N4 = (size_t)NTOT / 4;
  if (i < N4) {
    out4[i]          = q4[i];
    out4[N4 + i]     = k4[i];
    out4[2 * N4 + i] = v4[i];
    uchar4 m = m4[i];
    float4 mf;
    mf.x = m.x ? 1.0f : 0.0f;
    mf.y = m.y ? 1.0f : 0.0f;
    mf.z = m.z ? 1.0f : 0.0f;
    mf.w = m.w ? 1.0f : 0.0f;
    out4[3 * N4 + i] = mf;
  }
}

// ---------------------------------------------------------------------------
// Kernel 2: per-head attention. One block = one (b,h) head. 8 waves of 32.
// LDS: Q,K,V tiles (16KB each) + 64x64 score/prob matrix (16KB) = 64KB.
// Staging uses gfx1250 async global->LDS DMA (ASYNCcnt) when available.
// ---------------------------------------------------------------------------
__global__ __launch_bounds__(256) void attn_head_kernel(
    const float* __restrict__ q, const float* __restrict__ k,
    const float* __restrict__ v, const unsigned char* __restrict__ mask,
    float* __restrict__ out_weight,   // d_out + 4N  (B,H,S,S)
    float* __restrict__ out_o) {      // d_out + 5N  (B,H,S,D)
  __shared__ float Qs[Sn * Dn];
  __shared__ float Ks[Sn * Dn];
  __shared__ float Vs[Sn * Dn];
  __shared__ float Ps[Sn * Sn];   // scores -> probs (post-mask)

  const int head = blockIdx.x;                   // 0 .. B*H-1
  const size_t base = (size_t)head * (Sn * Dn);  // same stride for S*S since S==D
  const int tid  = threadIdx.x;
  const unsigned char* mrow_base = mask + base;

  // Pull the 4KB mask tile toward the WGP while we stage + run the score GEMM.
  if (tid < 64) __builtin_prefetch(mrow_base + tid * 64, 0, 0);

  // ---- stage Q,K,V into LDS ------------------------------------------------
#if USE_ASYNC_LDS
  {
    // Per-lane async DMA: 16B per lane per op, LDS written directly (no VGPR
    // round-trip). Tracked by ASYNCcnt.
    const uintptr_t gq = (uintptr_t)(q + base);
    const uintptr_t gk = (uintptr_t)(k + base);
    const uintptr_t gv = (uintptr_t)(v + base);
    const unsigned lq = (unsigned)(uintptr_t)Qs;   // low 32 bits = LDS offset
    const unsigned lk = (unsigned)(uintptr_t)Ks;
    const unsigned lv = (unsigned)(uintptr_t)Vs;
    for (int i = tid; i < (Sn * Dn) / 4; i += 256) {
      const unsigned boff = (unsigned)i * 16u;
      __builtin_amdgcn_global_load_async_to_lds_b128(
          (gv4i_p)(gq + boff), (lv4i_p)(uintptr_t)(lq + boff), 0, 0);
      __builtin_amdgcn_global_load_async_to_lds_b128(
          (gv4i_p)(gk + boff), (lv4i_p)(uintptr_t)(lk + boff), 0, 0);
      __builtin_amdgcn_global_load_async_to_lds_b128(
          (gv4i_p)(gv + boff), (lv4i_p)(uintptr_t)(lv + boff), 0, 0);
    }
#if __has_builtin(__builtin_amdgcn_s_wait_asynccnt)
    __builtin_amdgcn_s_wait_asynccnt(0);
#else
    asm volatile("s_wait_asynccnt 0" ::: "memory");
#endif
  }
#else
  {
    const float4* q4 = (const float4*)(q + base);
    const float4* k4 = (const float4*)(k + base);
    const float4* v4 = (const float4*)(v + base);
    float4* Qs4 = (float4*)Qs;
    float4* Ks4 = (float4*)Ks;
    float4* Vs4 = (float4*)Vs;
    for (int i = tid; i < (Sn * Dn) / 4; i += 256) {
      Qs4[i] = q4[i];
      Ks4[i] = k4[i];
      Vs4[i] = v4[i];
    }
  }
#endif
  __syncthreads();

  const int wave  = tid >> 5;       // 0..7
  const int lane  = tid & 31;
  const int lhalf = lane >> 4;      // 0: lanes 0-15, 1: lanes 16-31
  const int l16   = lane & 15;

  // ---- Phase 1: scores = Q @ K^T, 16x16 tiles, two tiles per wave ----------
  for (int t = wave * 2; t < wave * 2 + 2; ++t) {
    const int mi = (t >> 2) * 16;   // tile row origin
    const int ni = (t & 3) * 16;    // tile col origin
    v8f c = {};
    for (int kk = 0; kk < Dn; kk += 4) {
      const int ka = kk + lhalf * 2;
      v2f a, b;
      // A = Q[mi:mi+16, kk:kk+4]  (16x4, M=l16, K per lane-half)
      a.x = Qs[(mi + l16) * Dn + ka];
      a.y = Qs[(mi + l16) * Dn + ka + 1];
      // B = K^T[kk:kk+4, ni:ni+16] i.e. B[kd][n] = K[ni+n][kd]
      b.x = Ks[(ni + l16) * Dn + ka];
      b.y = Ks[(ni + l16) * Dn + ka + 1];
      c = __builtin_amdgcn_wmma_f32_16x16x4_f32(
          /*neg_a=*/false, a, /*neg_b=*/false, b,
          /*c_mod=*/(short)0, c, /*reuse_a=*/false, /*reuse_b=*/false);
    }
    // C/D layout: VGPR r -> M = r + 8*lhalf, N = l16
#pragma unroll
    for (int r = 0; r < 8; ++r) {
      const int row = mi + r + lhalf * 8;
      Ps[row * Sn + (ni + l16)] = c[r];
    }
  }
  __syncthreads();

  // ---- Phase 2: row softmax, then post-softmax multiplicative mask ---------
  if (tid < Sn) {
    float* row = Ps + tid * Sn;
    float mx = -INFINITY;
    for (int j = 0; j < Sn; ++j) mx = fmaxf(mx, row[j]);
    float sum = 0.0f;
    for (int j = 0; j < Sn; ++j) {
      float e = __expf(row[j] - mx);
      row[j] = e;
      sum += e;
    }
    const float inv = 1.0f / sum;
    const unsigned char* mrow = mrow_base + tid * Sn;
    float* wout = out_weight + base + (size_t)tid * Sn;
    for (int j = 0; j < Sn; ++j) {
      const float p = row[j] * inv * (mrow[j] ? 1.0f : 0.0f);
      row[j] = p;       // keep masked probs for P @ V
      wout[j] = p;      // attn_weight output (post-mask, per reference)
    }
  }
  __syncthreads();

  // ---- Phase 3: O = P @ V, second mask multiply, store ---------------------
  for (int t = wave * 2; t < wave * 2 + 2; ++t) {
    const int mi = (t >> 2) * 16;
    const int ni = (t & 3) * 16;
    v8f c = {};
    for (int kk = 0; kk < Sn; kk += 4) {
      const int ka = kk + lhalf * 2;
      v2f a, b;
      // A = P[mi:mi+16, kk:kk+4]
      a.x = Ps[(mi + l16) * Sn + ka];
      a.y = Ps[(mi + l16) * Sn + ka + 1];
      // B = V[kk:kk+4, ni:ni+16]  (already K x N)
      b.x = Vs[ka * Dn + (ni + l16)];
      b.y = Vs[(ka + 1) * Dn + (ni + l16)];
      c = __builtin_amdgcn_wmma_f32_16x16x4_f32(
          false, a, false, b, (short)0, c, false, false);
    }
#pragma unroll
    for (int r = 0; r < 8; ++r) {
      const int row = mi + r + lhalf * 8;
      const int col = ni + l16;
      const float mf = mrow_base[row * Sn + col] ? 1.0f : 0.0f;
      out_o[base + (size_t)row * Dn + col] = c[r] * mf;
    }
  }
}

// ---------------------------------------------------------------------------
extern "C" void kernel_launch(void* const* d_in, const int* in_sizes, int n_in,
                              void* d_out, int out_size, void* d_ws, size_t ws_size,
                              hipStream_t stream) {
  const float* q = (const float*)d_in[0];
  const float* k = (const float*)d_in[1];
  const float* v = (const float*)d_in[2];
  const unsigned char* m = (const unsigned char*)d_in[3];  // bool mask
  float* out = (float*)d_out;
  const size_t N = (size_t)NTOT;

  copy_passthru_kernel<<<(NTOT / 4 + 255) / 256, 256, 0, stream>>>(
      (const float4*)q, (const float4*)k, (const float4*)v,
      (const uchar4*)m, (float4*)out);
  attn_head_kernel<<<Bn * Hn, 256, 0, stream>>>(q, k, v, m,
                                                out + 4 * N, out + 5 * N);
}